// modal_knn_filling_31791347925428
// MI455X (gfx1250) — compile-verified
//
#include <hip/hip_runtime.h>
#include <math.h>
#include <stdint.h>

#ifndef __has_builtin
#define __has_builtin(x) 0
#endif

typedef __attribute__((ext_vector_type(2))) float v2f;
typedef __attribute__((ext_vector_type(8))) float v8f;
typedef int async_v4i __attribute__((vector_size(16)));   // matches builtin param

#define TILE_M 128
#define TILE_N 128
#define TILE_K 32
#define APAD 4   // row stride 36 floats: 16B-aligned rows, conflict-free a-frag reads
#define BPAD 4   // row stride 132 floats: 16B-aligned rows

#if __has_builtin(__builtin_amdgcn_global_load_async_to_lds_b128)
#define HAVE_ASYNC_LDS 1
#else
#define HAVE_ASYNC_LDS 0
#endif

// 16-byte global -> LDS copy. Async (ASYNCcnt, no VGPR round-trip) when the
// gfx1250 builtin is available; synchronous float4 staging otherwise.
__device__ __forceinline__ void copy16_to_lds(const float* g, float* l)
{
#if HAVE_ASYNC_LDS
    __builtin_amdgcn_global_load_async_to_lds_b128(
        (__attribute__((address_space(1))) async_v4i*)(uintptr_t)g,
        (__attribute__((address_space(3))) async_v4i*)(uint32_t)(uintptr_t)l,
        0, 0);
#else
    *reinterpret_cast<float4*>(l) = *reinterpret_cast<const float4*>(g);
#endif
}

__device__ __forceinline__ void wait_async_copies()
{
#if HAVE_ASYNC_LDS
#if __has_builtin(__builtin_amdgcn_s_wait_asynccnt)
    __builtin_amdgcn_s_wait_asynccnt(0);
#else
    asm volatile("s_wait_asynccnt 0x0" ::: "memory");
#endif
#endif
}

// Stage one 128x32 A tile and one 32x128 B tile into LDS (256 threads).
__device__ __forceinline__ void stage_tile(
    const float* __restrict__ A, const float* __restrict__ B,
    int K, int N, int blockM, int blockN, int k0, int tid,
    float (&As)[TILE_M][TILE_K + APAD], float (&Bs)[TILE_K][TILE_N + BPAD])
{
    #pragma unroll
    for (int it = 0; it < 4; ++it) {
        int idx = tid + it * 256;
        int r = idx >> 3, c4 = (idx & 7) << 2;
        copy16_to_lds(&A[(size_t)(blockM + r) * K + k0 + c4], &As[r][c4]);
    }
    #pragma unroll
    for (int it = 0; it < 4; ++it) {
        int idx = tid + it * 256;
        int r = idx >> 5, c4 = (idx & 31) << 2;
        copy16_to_lds(&B[(size_t)(k0 + r) * N + blockN + c4], &Bs[r][c4]);
    }
}

// ---------------------------------------------------------------------------
// Tiled fp32 GEMM on the WMMA pipe: C[M,N] = act(A[M,K] @ B[K,N] + bias)
// Block = 256 threads (8 waves). Wave (waveM,waveN) owns a 32x64 sub-tile
// = 2x4 V_WMMA_F32_16X16X4_F32 tiles. Double-buffered LDS staging (async
// global->LDS when available). Requires M%128==0, N%128==0, K%32==0.
// ---------------------------------------------------------------------------
__global__ __launch_bounds__(256)
void wmma_gemm_bias_act(const float* __restrict__ A, const float* __restrict__ B,
                        const float* __restrict__ bias, float* __restrict__ C,
                        int M, int N, int K, int relu)
{
    __shared__ float As[2][TILE_M][TILE_K + APAD];   // 2 x 18 KB
    __shared__ float Bs[2][TILE_K][TILE_N + BPAD];   // 2 x 16.5 KB

    const int tid   = threadIdx.x;
    const int lane  = tid & 31;
    const int wave  = tid >> 5;      // 0..7
    const int waveM = wave >> 1;     // 0..3 -> 32-row slab
    const int waveN = wave & 1;      // 0..1 -> 64-col slab

    const int blockN = blockIdx.x * TILE_N;
    const int blockM = blockIdx.y * TILE_M;

    const int mrow = lane & 15;          // M (a-frag) / N (b-frag) within tile
    const int kb   = (lane >> 4) << 1;   // lane-half selects K pair {0,1} or {2,3}

    v8f zero = {0.f, 0.f, 0.f, 0.f, 0.f, 0.f, 0.f, 0.f};
    v8f acc[2][4];
    #pragma unroll
    for (int i = 0; i < 2; ++i)
        #pragma unroll
        for (int j = 0; j < 4; ++j)
            acc[i][j] = zero;

    const int nk = K / TILE_K;

    stage_tile(A, B, K, N, blockM, blockN, 0, tid, As[0], Bs[0]);
    wait_async_copies();
    __syncthreads();

    int buf = 0;
    for (int kt = 0; kt < nk; ++kt) {
        if (kt + 1 < nk)
            stage_tile(A, B, K, N, blockM, blockN, (kt + 1) * TILE_K, tid,
                       As[buf ^ 1], Bs[buf ^ 1]);

        #pragma unroll
        for (int kk = 0; kk < TILE_K; kk += 4) {
            v2f afrag[2];
            #pragma unroll
            for (int i = 0; i < 2; ++i) {
                const int r = waveM * 32 + i * 16 + mrow;
                afrag[i] = *reinterpret_cast<const v2f*>(&As[buf][r][kk + kb]);
            }
            v2f bfrag[4];
            #pragma unroll
            for (int j = 0; j < 4; ++j) {
                const int c = waveN * 64 + j * 16 + mrow;
                bfrag[j].x = Bs[buf][kk + kb][c];
                bfrag[j].y = Bs[buf][kk + kb + 1][c];
            }
            #pragma unroll
            for (int i = 0; i < 2; ++i)
                #pragma unroll
                for (int j = 0; j < 4; ++j)
                    acc[i][j] = __builtin_amdgcn_wmma_f32_16x16x4_f32(
                        false, afrag[i], false, bfrag[j],
                        (short)0, acc[i][j], false, false);
        }

        wait_async_copies();
        __syncthreads();
        buf ^= 1;
    }

    // Epilogue: C/D layout -> lanes 0-15: M=v, lanes 16-31: M=8+v; N = lane&15
    const int rOff = (lane >> 4) << 3;   // 0 or 8
    #pragma unroll
    for (int i = 0; i < 2; ++i) {
        const int rowBase = blockM + waveM * 32 + i * 16 + rOff;
        #pragma unroll
        for (int j = 0; j < 4; ++j) {
            const int col = blockN + waveN * 64 + j * 16 + mrow;
            const float bv = bias ? bias[col] : 0.0f;
            #pragma unroll
            for (int v = 0; v < 8; ++v) {
                float val = acc[i][j][v] + bv;
                if (relu) val = fmaxf(val, 0.0f);
                C[(size_t)(rowBase + v) * N + col] = val;
            }
        }
    }
}

// ---------------------------------------------------------------------------
// Row L2 norms: one block (128 thr) per row of [rows, F]
// ---------------------------------------------------------------------------
__global__ __launch_bounds__(128)
void rownorm_kernel(const float* __restrict__ P, float* __restrict__ norms, int F)
{
    __shared__ float red[128];
    const int r = blockIdx.x, tid = threadIdx.x;
    float s = 0.f;
    for (int c = tid; c < F; c += 128) {
        float x = P[(size_t)r * F + c];
        s += x * x;
    }
    red[tid] = s;
    __syncthreads();
    for (int off = 64; off > 0; off >>= 1) {
        if (tid < off) red[tid] += red[tid + off];
        __syncthreads();
    }
    if (tid == 0) norms[r] = sqrtf(red[0]);
}

// ---------------------------------------------------------------------------
// Streaming top-3 cosine similarity. One block per row; only missing rows do
// work (available rows never consume their top-k in the reference).
// Candidates exclude missing columns. Per-thread top-3, merged via LDS.
// ---------------------------------------------------------------------------
__global__ __launch_bounds__(256)
void topk3_kernel(const float* __restrict__ P, const float* __restrict__ norms,
                  const int* __restrict__ missing, int modalId,
                  float* __restrict__ topv, int* __restrict__ topi, int Bn, int F)
{
    __shared__ float prow[512];
    __shared__ float sv[768];
    __shared__ int   si[768];

    const int r = blockIdx.x;
    if (missing[r] != modalId) return;   // uniform branch: whole block exits

    const int tid = threadIdx.x;
    for (int c = tid; c < F; c += 256) prow[c] = P[(size_t)r * F + c];
    __syncthreads();

    const float nr = norms[r];
    const float NEG = -3.402823466e38f;
    float v0 = NEG, v1 = NEG, v2 = NEG;
    int   i0 = 0,   i1 = 0,   i2 = 0;

    for (int c = tid; c < Bn; c += 256) {
        if (missing[c] == modalId) continue;   // masked (missing) column
        const float* pc = &P[(size_t)c * F];
        float dot = 0.f;
        #pragma unroll 4
        for (int k = 0; k < F; k += 4) {
            dot += prow[k + 0] * pc[k + 0];
            dot += prow[k + 1] * pc[k + 1];
            dot += prow[k + 2] * pc[k + 2];
            dot += prow[k + 3] * pc[k + 3];
        }
        const float den = fmaxf(nr * norms[c], 1e-8f);
        const float s = dot / den;
        if (s > v0)      { v2 = v1; i2 = i1; v1 = v0; i1 = i0; v0 = s; i0 = c; }
        else if (s > v1) { v2 = v1; i2 = i1; v1 = s;  i1 = c; }
        else if (s > v2) { v2 = s;  i2 = c; }
    }

    sv[tid * 3 + 0] = v0; si[tid * 3 + 0] = i0;
    sv[tid * 3 + 1] = v1; si[tid * 3 + 1] = i1;
    sv[tid * 3 + 2] = v2; si[tid * 3 + 2] = i2;
    __syncthreads();

    if (tid == 0) {
        float b0 = NEG, b1 = NEG, b2 = NEG;
        int   j0 = 0,   j1 = 0,   j2 = 0;
        for (int t = 0; t < 768; ++t) {
            const float s = sv[t];
            const int   c = si[t];
            if (s > b0)      { b2 = b1; j2 = j1; b1 = b0; j1 = j0; b0 = s; j0 = c; }
            else if (s > b1) { b2 = b1; j2 = j1; b1 = s;  j1 = c; }
            else if (s > b2) { b2 = s;  j2 = c; }
        }
        topv[r * 3 + 0] = b0; topi[r * 3 + 0] = j0;
        topv[r * 3 + 1] = b1; topi[r * 3 + 1] = j1;
        topv[r * 3 + 2] = b2; topi[r * 3 + 2] = j2;
    }
}

// ---------------------------------------------------------------------------
// Fill missing rows with softmax-weighted top-3 gather; copy others.
// Writes directly into the concatenated [B, 3F] activation.
// ---------------------------------------------------------------------------
__global__ __launch_bounds__(256)
void fill_concat_kernel(const float* __restrict__ P, const float* __restrict__ topv,
                        const int* __restrict__ topi, const int* __restrict__ missing,
                        int modalId, float* __restrict__ Xcat, int F, int ldx, int colOff)
{
    const int r = blockIdx.x, tid = threadIdx.x;
    if (missing[r] != modalId) {
        for (int c = tid; c < F; c += 256)
            Xcat[(size_t)r * ldx + colOff + c] = P[(size_t)r * F + c];
        return;
    }
    const float t0 = topv[r * 3 + 0], t1 = topv[r * 3 + 1], t2 = topv[r * 3 + 2];
    const int   j0 = topi[r * 3 + 0], j1 = topi[r * 3 + 1], j2 = topi[r * 3 + 2];
    const float m  = fmaxf(t0, fmaxf(t1, t2));
    const float e0 = expf(t0 - m), e1 = expf(t1 - m), e2 = expf(t2 - m);
    const float inv = 1.f / (e0 + e1 + e2);
    const float w0 = e0 * inv, w1 = e1 * inv, w2 = e2 * inv;
    for (int c = tid; c < F; c += 256)
        Xcat[(size_t)r * ldx + colOff + c] =
            w0 * P[(size_t)j0 * F + c] + w1 * P[(size_t)j1 * F + c] + w2 * P[(size_t)j2 * F + c];
}

// ---------------------------------------------------------------------------
// Final GEMV: out[r] = H[r,:] . W2 + b2, one wave32 per row, 8 rows / block.
// ---------------------------------------------------------------------------
__global__ __launch_bounds__(256)
void gemv_kernel(const float* __restrict__ H, const float* __restrict__ W2,
                 const float* __restrict__ b2, float* __restrict__ out, int F)
{
    const int lane = threadIdx.x & 31;
    const int wave = threadIdx.x >> 5;
    const int r = blockIdx.x * 8 + wave;
    float s = 0.f;
    for (int k = lane; k < F; k += 32) s += H[(size_t)r * F + k] * W2[k];
    for (int off = 16; off > 0; off >>= 1) s += __shfl_xor(s, off, 32);
    if (lane == 0) out[r] = s + b2[0];
}

// ---------------------------------------------------------------------------
extern "C" void kernel_launch(void* const* d_in, const int* in_sizes, int n_in,
                              void* d_out, int out_size, void* d_ws, size_t ws_size,
                              hipStream_t stream)
{
    const int Bn = 4096, D = 1024, F = 512;

    const float* X[3]  = { (const float*)d_in[0], (const float*)d_in[1], (const float*)d_in[2] };
    const int* missing = (const int*)d_in[3];
    const float* W[3]  = { (const float*)d_in[4], (const float*)d_in[6], (const float*)d_in[8] };
    const float* bW[3] = { (const float*)d_in[5], (const float*)d_in[7], (const float*)d_in[9] };
    const float* W1 = (const float*)d_in[10];
    const float* b1 = (const float*)d_in[11];
    const float* W2 = (const float*)d_in[12];
    const float* b2 = (const float*)d_in[13];
    float* out = (float*)d_out;

    // Workspace carve-out (~59 MB, all 16B-aligned offsets)
    char* ws = (char*)d_ws;
    float* P     = (float*)ws; ws += (size_t)3 * Bn * F * sizeof(float);   // 24 MB
    float* norms = (float*)ws; ws += (size_t)3 * Bn * sizeof(float);
    float* topv  = (float*)ws; ws += (size_t)3 * Bn * 3 * sizeof(float);
    int*   topi  = (int*)ws;   ws += (size_t)3 * Bn * 3 * sizeof(int);
    float* Xcat  = (float*)ws; ws += (size_t)Bn * 3 * F * sizeof(float);   // 24 MB
    float* H     = (float*)ws; ws += (size_t)Bn * F * sizeof(float);       // 8 MB
    (void)ws_size; (void)in_sizes; (void)n_in; (void)out_size;

    // 1) Projections: P_m = X_m @ W_m + b_m   (WMMA fp32)
    dim3 gProj(F / TILE_N, Bn / TILE_M);   // (4, 32)
    for (int m = 0; m < 3; ++m)
        wmma_gemm_bias_act<<<gProj, 256, 0, stream>>>(
            X[m], W[m], bW[m], P + (size_t)m * Bn * F, Bn, F, D, 0);

    // 2) Row norms for all 3 modalities at once (P is contiguous [3*B, F])
    rownorm_kernel<<<3 * Bn, 128, 0, stream>>>(P, norms, F);

    // 3) Top-3 cosine neighbors for missing rows, per modality
    for (int m = 0; m < 3; ++m)
        topk3_kernel<<<Bn, 256, 0, stream>>>(
            P + (size_t)m * Bn * F, norms + (size_t)m * Bn, missing, m + 1,
            topv + (size_t)m * Bn * 3, topi + (size_t)m * Bn * 3, Bn, F);

    // 4) KNN fill + concat into Xcat [B, 3F]
    for (int m = 0; m < 3; ++m)
        fill_concat_kernel<<<Bn, 256, 0, stream>>>(
            P + (size_t)m * Bn * F, topv + (size_t)m * Bn * 3,
            topi + (size_t)m * Bn * 3, missing, m + 1,
            Xcat, F, 3 * F, m * F);

    // 5) H = relu(Xcat @ W1 + b1)   (WMMA fp32, K=1536)
    dim3 gMlp(F / TILE_N, Bn / TILE_M);
    wmma_gemm_bias_act<<<gMlp, 256, 0, stream>>>(Xcat, W1, b1, H, Bn, F, 3 * F, 1);

    // 6) out = H @ W2 + b2
    gemv_kernel<<<Bn / 8, 256, 0, stream>>>(H, W2, b2, out, F);
}